// TtMambaBlock_68444598829166
// MI455X (gfx1250) — compile-verified
//
#include <hip/hip_runtime.h>
#include <hip/hip_bf16.h>
#include <math.h>

// ---------------------------------------------------------------------------
// Mamba decode step for MI455X (gfx1250), wave32, WMMA f32 16x16x4.
// Shapes: B=128, DM=2560, DI=5120, D_STATE=32, DT_RANK=160, XPROJ=224.
// Memory-bound (~240MB @ 23.3TB/s ~= 10us); fp32 WMMA preserves precision.
// ---------------------------------------------------------------------------

#define BATCH   128
#define DM      2560
#define DI      5120
#define DSTATE  32
#define DTRANK  160
#define NXPROJ  (DTRANK + 2 * DSTATE)   // 224

typedef __attribute__((ext_vector_type(2))) float v2f;
typedef __attribute__((ext_vector_type(8))) float v8f;

// D = A(16x4) * B(4x16) + C  per CDNA5 ISA 7.12.2:
//   A: lane[0..15] row M=lane, vgpr0/1 = K{0,1}; lane[16..31] row M=lane-16, K{2,3}
//   B: lane[0..15] col N=lane, vgpr0/1 = K{0,1}; lane[16..31] col N=lane-16, K{2,3}
//   C/D: vgpr v, lane l -> row = (l>>4)*8 + v, col = l&15
__device__ __forceinline__ v8f wmma_f32(v2f a, v2f b, v8f c) {
    return __builtin_amdgcn_wmma_f32_16x16x4_f32(false, a, false, b,
                                                 (short)0, c, false, false);
}

// Fast SiLU: x * rcp(1+e^-x)  ->  v_exp_f32 + v_rcp_f32 (no IEEE div expansion)
__device__ __forceinline__ float silu(float x) {
    return x * __builtin_amdgcn_rcpf(1.0f + __expf(-x));
}

// Fast stable softplus -> v_exp_f32 + v_log_f32
__device__ __forceinline__ float softplus(float t) {
    return (t > 20.0f) ? t : __logf(1.0f + __expf(t));
}

// ---------------------------------------------------------------------------
// Kernel 1: fused dual input projection + conv tap + both SiLUs.
//   residual = silu(X @ Wmlp)
//   u        = silu(conv_w[0..2].cs[1..3] + conv_w[3].(X @ Wssm) + conv_b)
// One wave computes a 64(M) x 16(N) strip for BOTH weight matrices so each
// A-fragment is loaded once and reused 8x (4 M-tiles x 2 matrices).
// x_ssm never touches memory: it exists only in the accumulators.
// grid = (DI/64, BATCH/64), block = 128 (4 waves, each a distinct N tile).
// ---------------------------------------------------------------------------
__global__ __launch_bounds__(128) void k_in_proj_dual(
    const float* __restrict__ X,             // [128,2560]
    const float* __restrict__ Wssm,          // [2560,5120]
    const float* __restrict__ Wmlp,          // [2560,5120]
    const float* __restrict__ conv_states,   // [4,128,5120]
    const float* __restrict__ conv_w,        // [4,5120]
    const float* __restrict__ conv_b,        // [5120]
    float* __restrict__ u,                   // [128,5120]
    float* __restrict__ residual)            // [128,5120]
{
    const int lane = threadIdx.x & 31;
    const int wave = threadIdx.x >> 5;
    const int col0 = blockIdx.x * 64 + wave * 16;
    const int row0 = blockIdx.y * 64;
    const int m    = lane & 15;
    const int kh   = lane >> 4;          // selects K pair {0,1} vs {2,3}
    const int col  = col0 + m;

    v8f acc_s[4] = {}, acc_m[4] = {};

    // A fragments indexed in float2 units -> saddr + voffset scale_offset form
    const v2f* X2 = (const v2f*)X;
    int aoff[4];
#pragma unroll
    for (int i = 0; i < 4; ++i)
        aoff[i] = (row0 + i * 16 + m) * (DM / 2) + kh;
    int woff = (kh * 2) * DI + col;      // float units for b32 loads

#define K1_STEP(PF)                                                        \
    {                                                                      \
        v2f bs, bm;                                                        \
        bs.x = Wssm[woff];                                                 \
        bs.y = Wssm[woff + DI];                                            \
        bm.x = Wmlp[woff];                                                 \
        bm.y = Wmlp[woff + DI];                                            \
        if (PF) {                                                          \
            __builtin_prefetch(&Wssm[woff + 128 * DI], 0, 1);              \
            __builtin_prefetch(&Wmlp[woff + 128 * DI], 0, 1);              \
        }                                                                  \
        _Pragma("unroll")                                                  \
        for (int i = 0; i < 4; ++i) {                                      \
            v2f a = X2[aoff[i]];                                           \
            acc_s[i] = wmma_f32(a, bs, acc_s[i]);                          \
            acc_m[i] = wmma_f32(a, bm, acc_m[i]);                          \
            aoff[i] += 2;                                                  \
        }                                                                  \
        woff += 4 * DI;                                                    \
    }

#pragma unroll 2
    for (int k = 0; k < DM - 128; k += 4) K1_STEP(true)
#pragma unroll 2
    for (int k = DM - 128; k < DM; k += 4) K1_STEP(false)
#undef K1_STEP

    // conv operands are per-lane invariant (column is fixed)
    const float cw0 = conv_w[col];
    const float cw1 = conv_w[DI + col];
    const float cw2 = conv_w[2 * DI + col];
    const float cw3 = conv_w[3 * DI + col];
    const float cb  = conv_b[col];
    const size_t N  = (size_t)BATCH * DI;

#pragma unroll
    for (int i = 0; i < 4; ++i) {
#pragma unroll
        for (int v = 0; v < 8; ++v) {
            const size_t idx = (size_t)(row0 + i * 16 + kh * 8 + v) * DI + col;
            residual[idx] = silu(acc_m[i][v]);
            const float conv_out = conv_states[N + idx]     * cw0
                                 + conv_states[2 * N + idx] * cw1
                                 + conv_states[3 * N + idx] * cw2
                                 + acc_s[i][v]              * cw3
                                 + cb;
            u[idx] = silu(conv_out);
        }
    }
}

// ---------------------------------------------------------------------------
// Kernel 2: dbl = u @ x_proj_w  [128,5120]x[5120,224].
// grid = (NXPROJ/16, BATCH/64), block = 128; wave -> one 16x16 tile.
// ---------------------------------------------------------------------------
__global__ __launch_bounds__(128) void k_x_proj(
    const float* __restrict__ u,
    const float* __restrict__ W,             // [5120,224]
    float* __restrict__ dbl)                 // [128,224]
{
    const int lane = threadIdx.x & 31;
    const int wave = threadIdx.x >> 5;
    const int row0 = (blockIdx.y * 4 + wave) * 16;
    const int col0 = blockIdx.x * 16;
    const int m    = lane & 15;
    const int kh   = lane >> 4;

    v8f acc = {};
    const v2f* u2 = (const v2f*)u;
    int aoff = (row0 + m) * (DI / 2) + kh;
    int woff = (kh * 2) * NXPROJ + col0 + m;
#pragma unroll 2
    for (int k = 0; k < DI; k += 4) {
        v2f a = u2[aoff];
        v2f b;
        b.x = W[woff];
        b.y = W[woff + NXPROJ];
        acc = wmma_f32(a, b, acc);
        aoff += 2;
        woff += 4 * NXPROJ;
    }
#pragma unroll
    for (int v = 0; v < 8; ++v)
        dbl[(size_t)(row0 + kh * 8 + v) * NXPROJ + col0 + m] = acc[v];
}

// ---------------------------------------------------------------------------
// Kernel 3: dt = softplus(dbl[:, :160] @ dt_proj_w + dt_proj_b)
//   [128,160]x[160,5120].  grid = (DI/64, BATCH/16), block = 128.
// ---------------------------------------------------------------------------
__global__ __launch_bounds__(128) void k_dt_proj(
    const float* __restrict__ dbl,           // [128,224] (use cols 0..159)
    const float* __restrict__ W,             // [160,5120]
    const float* __restrict__ bias,          // [5120]
    float* __restrict__ dt)                  // [128,5120]
{
    const int lane = threadIdx.x & 31;
    const int wave = threadIdx.x >> 5;
    const int col0 = blockIdx.x * 64 + wave * 16;
    const int row0 = blockIdx.y * 16;
    const int m    = lane & 15;
    const int kh   = lane >> 4;

    v8f acc = {};
    const v2f* dbl2 = (const v2f*)dbl;
    int aoff = (row0 + m) * (NXPROJ / 2) + kh;
    int woff = (kh * 2) * DI + col0 + m;
#pragma unroll 2
    for (int k = 0; k < DTRANK; k += 4) {
        v2f a = dbl2[aoff];
        v2f b;
        b.x = W[woff];
        b.y = W[woff + DI];
        acc = wmma_f32(a, b, acc);
        aoff += 2;
        woff += 4 * DI;
    }
    const float bc = bias[col0 + m];
#pragma unroll
    for (int v = 0; v < 8; ++v)
        dt[(size_t)(row0 + kh * 8 + v) * DI + col0 + m] = softplus(acc[v] + bc);
}

// ---------------------------------------------------------------------------
// Kernel 4: SSM recurrence + contraction + residual gate.
//   z[b,d] = (sum_s (exp(dt*(-exp(A_log)))*h + dt*Bm[s]*u) * Cm[s] + D*u) * residual
// One thread per (b,d); ssm_state row (32 floats) is a contiguous 128B read.
// Bm/Cm (per-batch, 64 floats) staged in LDS.  grid = (DI/256, BATCH).
// ---------------------------------------------------------------------------
__global__ __launch_bounds__(256) void k_ssm(
    const float* __restrict__ ssm_state,     // [128,5120,32]
    const float* __restrict__ A_log,         // [5120,32]
    const float* __restrict__ dbl,           // [128,224]
    const float* __restrict__ dt,            // [128,5120]
    const float* __restrict__ u,             // [128,5120]
    const float* __restrict__ Dvec,          // [5120]
    const float* __restrict__ residual,      // [128,5120]
    float* __restrict__ z)                   // [128,5120]
{
    __shared__ float sB[DSTATE];
    __shared__ float sC[DSTATE];
    const int b = blockIdx.y;
    const int d = blockIdx.x * 256 + threadIdx.x;
    if (threadIdx.x < DSTATE)
        sB[threadIdx.x] = dbl[(size_t)b * NXPROJ + DTRANK + threadIdx.x];
    else if (threadIdx.x < 2 * DSTATE)
        sC[threadIdx.x - DSTATE] = dbl[(size_t)b * NXPROJ + DTRANK + DSTATE + (threadIdx.x - DSTATE)];
    __syncthreads();

    const size_t bd = (size_t)b * DI + d;
    const float dtv = dt[bd];
    const float uv  = u[bd];
    const float* hs = ssm_state + bd * DSTATE;
    const float* al = A_log + (size_t)d * DSTATE;

    float y = 0.0f;
#pragma unroll
    for (int s = 0; s < DSTATE; ++s) {
        const float A  = -__expf(al[s]);
        const float dA = __expf(dtv * A);
        const float h  = dA * hs[s] + dtv * sB[s] * uv;
        y += h * sC[s];
    }
    y += Dvec[d] * uv;
    z[bd] = y * residual[bd];
}

// ---------------------------------------------------------------------------
// Kernel 5: out = z @ out_proj_w  [128,5120]x[5120,2560].
// Same 64x16 per-wave strip as kernel 1.  grid = (DM/64, BATCH/64), block 128.
// ---------------------------------------------------------------------------
__global__ __launch_bounds__(128) void k_out_proj(
    const float* __restrict__ z,
    const float* __restrict__ W,             // [5120,2560]
    float* __restrict__ out)                 // [128,2560]
{
    const int lane = threadIdx.x & 31;
    const int wave = threadIdx.x >> 5;
    const int col0 = blockIdx.x * 64 + wave * 16;
    const int row0 = blockIdx.y * 64;
    const int m    = lane & 15;
    const int kh   = lane >> 4;

    v8f acc[4] = {};
    const v2f* z2 = (const v2f*)z;
    int aoff[4];
#pragma unroll
    for (int i = 0; i < 4; ++i)
        aoff[i] = (row0 + i * 16 + m) * (DI / 2) + kh;
    int woff = (kh * 2) * DM + col0 + m;

#define K5_STEP(PF)                                                        \
    {                                                                      \
        v2f b;                                                             \
        b.x = W[woff];                                                     \
        b.y = W[woff + DM];                                                \
        if (PF)                                                            \
            __builtin_prefetch(&W[woff + 128 * DM], 0, 1);                 \
        _Pragma("unroll")                                                  \
        for (int i = 0; i < 4; ++i) {                                      \
            v2f a = z2[aoff[i]];                                           \
            acc[i] = wmma_f32(a, b, acc[i]);                               \
            aoff[i] += 2;                                                  \
        }                                                                  \
        woff += 4 * DM;                                                    \
    }

#pragma unroll 2
    for (int k = 0; k < DI - 128; k += 4) K5_STEP(true)
#pragma unroll 2
    for (int k = DI - 128; k < DI; k += 4) K5_STEP(false)
#undef K5_STEP

#pragma unroll
    for (int i = 0; i < 4; ++i)
#pragma unroll
        for (int v = 0; v < 8; ++v)
            out[(size_t)(row0 + i * 16 + kh * 8 + v) * DM + col0 + m] = acc[i][v];
}

// ---------------------------------------------------------------------------
extern "C" void kernel_launch(void* const* d_in, const int* in_sizes, int n_in,
                              void* d_out, int out_size, void* d_ws, size_t ws_size,
                              hipStream_t stream) {
    const float* x           = (const float*)d_in[0];   // [1,1,128,2560]
    const float* conv_states = (const float*)d_in[1];   // [4,128,5120]
    const float* ssm_state   = (const float*)d_in[2];   // [128,5120,32]
    const float* w_in_ssm    = (const float*)d_in[3];   // [2560,5120]
    const float* w_in_mlp    = (const float*)d_in[4];   // [2560,5120]
    const float* conv_w      = (const float*)d_in[5];   // [4,5120]
    const float* conv_b      = (const float*)d_in[6];   // [5120]
    const float* A_log       = (const float*)d_in[7];   // [5120,32]
    const float* x_proj_w    = (const float*)d_in[8];   // [5120,224]
    const float* dt_proj_w   = (const float*)d_in[9];   // [160,5120]
    const float* dt_proj_b   = (const float*)d_in[10];  // [5120]
    const float* Dvec        = (const float*)d_in[11];  // [5120]
    const float* out_proj_w  = (const float*)d_in[12];  // [5120,2560]
    float* out = (float*)d_out;                          // [128,2560]

    float* ws = (float*)d_ws;
    const size_t NBD = (size_t)BATCH * DI;               // 655360
    float* resid = ws;
    float* ubuf  = ws + NBD;
    float* dtbuf = ws + 2 * NBD;
    float* zbuf  = ws + 3 * NBD;
    float* dbl   = ws + 4 * NBD;                         // 128*224

    k_in_proj_dual<<<dim3(DI / 64, BATCH / 64), 128, 0, stream>>>(
        x, w_in_ssm, w_in_mlp, conv_states, conv_w, conv_b, ubuf, resid);

    k_x_proj<<<dim3(NXPROJ / 16, BATCH / 64), 128, 0, stream>>>(
        ubuf, x_proj_w, dbl);

    k_dt_proj<<<dim3(DI / 64, BATCH / 16), 128, 0, stream>>>(
        dbl, dt_proj_w, dt_proj_b, dtbuf);

    k_ssm<<<dim3(DI / 256, BATCH), 256, 0, stream>>>(
        ssm_state, A_log, dbl, dtbuf, ubuf, Dvec, resid, zbuf);

    k_out_proj<<<dim3(DM / 64, BATCH / 64), 128, 0, stream>>>(
        zbuf, out_proj_w, out);
}